// QModel_65481071397010
// MI455X (gfx1250) — compile-verified
//
#include <hip/hip_runtime.h>
#include <cstdint>
#include <cstddef>

#define NUM_WIRES 14
#define DIMS      (1 << NUM_WIRES)   // 16384
#define PAIRS     (DIMS / 2)         // 8192
#define NUM_LAYERS 6
#define NTHREADS  512
#define NWAVES    (NTHREADS / 32)    // 16
#define NTILES    (DIMS / 16 / 16)   // 64 column tiles of 16

typedef __attribute__((ext_vector_type(4))) unsigned int v4u;
typedef __attribute__((ext_vector_type(8))) int v8i;
typedef __attribute__((ext_vector_type(4))) int v4i;
typedef __attribute__((ext_vector_type(2))) float v2f;
typedef __attribute__((ext_vector_type(8))) float v8f;

__device__ __forceinline__ unsigned insert_bit(unsigned v, int pos, unsigned b) {
  unsigned mask = (1u << pos) - 1u;
  return ((v & ~mask) << 1) | (b << pos) | (v & mask);
}

__global__ __launch_bounds__(NTHREADS)
void qsim_kernel(const float* __restrict__ state,   // (B, DIM) real fp32
                 const float* __restrict__ params,  // (L, 2n)
                 const float* __restrict__ head_w,  // (1, n)
                 const float* __restrict__ head_b,  // (1,)
                 float* __restrict__ out)           // (B,)
{
  __shared__ __align__(16) float s_re[DIMS];   // 64 KB
  __shared__ __align__(16) float s_im[DIMS];   // 64 KB
  __shared__ float s_red[NUM_WIRES];

  const int tid  = threadIdx.x;
  const int lane = tid & 31;
  const int wv   = tid >> 5;
  const int b    = blockIdx.x;
  const float* g = state + (size_t)b * DIMS;

  // ---- Load real part of the state into LDS via one TDM DMA ----
#if __has_builtin(__builtin_amdgcn_tensor_load_to_lds)
  if (tid < 32) {  // wave 0 issues the TDM op (EXEC ignored by TDM)
    const uint64_t ga  = (uint64_t)(uintptr_t)g;
    const unsigned lds = (unsigned)(uintptr_t)(void*)&s_re[0];
    v4u g0;
    g0.x = 1u;                                                // count = 1 valid descriptor
    g0.y = lds;                                               // lds_addr
    g0.z = (unsigned)(ga & 0xFFFFFFFFu);                      // global_addr[31:0]
    g0.w = (unsigned)((ga >> 32) & 0x1FFFFFFu) | (2u << 30);  // addr[56:32] | type=2
    v8i g1;
    g1[0] = (int)(2u << 16);                            // wg_mask=0, data_size=4B
    g1[1] = (int)(((unsigned)DIMS & 0xFFFFu) << 16);    // tensor_dim0[15:0]
    g1[2] = (int)(((unsigned)DIMS >> 16) | (1u << 16)); // dim0[31:16], tensor_dim1=1 lo
    g1[3] = (int)((unsigned)DIMS << 16);                // tensor_dim1 hi=0, tile_dim0=DIMS
    g1[4] = 1;                                          // tile_dim1=1, tile_dim2=0
    g1[5] = DIMS;                                       // tensor_dim0_stride lo32
    g1[6] = 0;
    g1[7] = 0;
    v4i z4 = {0, 0, 0, 0};
    v8i z8 = {0, 0, 0, 0, 0, 0, 0, 0};
    __builtin_amdgcn_tensor_load_to_lds(g0, g1, z4, z4, z8, 0);
#if __has_builtin(__builtin_amdgcn_s_wait_tensorcnt)
    __builtin_amdgcn_s_wait_tensorcnt(0);
#else
    asm volatile("s_wait_tensorcnt 0" ::: "memory");
#endif
  }
#else
  for (int k = tid; k < DIMS / 4; k += NTHREADS)
    ((float4*)s_re)[k] = ((const float4*)g)[k];
#endif
  // zero imaginary part
  float4 zz; zz.x = zz.y = zz.z = zz.w = 0.f;
  for (int k = tid; k < DIMS / 4; k += NTHREADS)
    ((float4*)s_im)[k] = zz;
  __syncthreads();

#pragma unroll 1
  for (int layer = 0; layer < NUM_LAYERS; ++layer) {
#if __has_builtin(__builtin_amdgcn_wmma_f32_16x16x4_f32)
    // ---- 4 fused 16x16 complex unitaries per layer, applied with WMMA ----
#pragma unroll 1
    for (int grp = 0; grp < 4; ++grp) {
      const int pp = (grp < 3) ? (10 - 4 * grp) : 0;  // low bit pos of 4-bit group
      // per-slot 2x2 unitaries (slot 0 = MSB of the 4-bit group index)
      float u_r[4][2][2], u_i[4][2][2];
      for (int s4 = 0; s4 < 4; ++s4) {
        int w = (grp < 3) ? (4 * grp + s4) : ((s4 < 2) ? -1 : 12 + (s4 - 2));
        if (w < 0) {  // identity lift
          u_r[s4][0][0] = 1.f; u_r[s4][0][1] = 0.f;
          u_r[s4][1][0] = 0.f; u_r[s4][1][1] = 1.f;
          u_i[s4][0][0] = u_i[s4][0][1] = u_i[s4][1][0] = u_i[s4][1][1] = 0.f;
        } else {
          const float th = params[layer * 2 * NUM_WIRES + w];
          const float ph = params[layer * 2 * NUM_WIRES + NUM_WIRES + w];
          float sn, cs, sp, cp;
          __sincosf(th * 0.5f, &sn, &cs);
          __sincosf(ph * 0.5f, &sp, &cp);
          // U = RZ(ph) @ RY(th)
          u_r[s4][0][0] =  cs * cp; u_i[s4][0][0] = -cs * sp;
          u_r[s4][0][1] = -sn * cp; u_i[s4][0][1] =  sn * sp;
          u_r[s4][1][0] =  sn * cp; u_i[s4][1][0] =  sn * sp;
          u_r[s4][1][1] =  cs * cp; u_i[s4][1][1] =  cs * sp;
        }
      }
      // Build per-lane A operand entries: M[m][k], k per 16x4 f32 A layout
      // (VGPR0 -> K = 4c + 2h, VGPR1 -> K = 4c + 2h + 1, h = lane>=16, m = lane&15)
      const int m = lane & 15, h = lane >> 4;
      const int m3 = (m >> 3) & 1, m2 = (m >> 2) & 1, m1 = (m >> 1) & 1, m0 = m & 1;
      v2f ar[4], ai[4];
#pragma unroll
      for (int c = 0; c < 4; ++c) {
#pragma unroll
        for (int v = 0; v < 2; ++v) {
          const int k = 4 * c + 2 * h + v;
          const int k3 = (k >> 3) & 1, k2 = (k >> 2) & 1, k1 = (k >> 1) & 1, k0 = k & 1;
          float pr = u_r[0][m3][k3], pi = u_i[0][m3][k3];
          float qr = u_r[1][m2][k2], qi = u_i[1][m2][k2];
          float t  = pr * qr - pi * qi; pi = pr * qi + pi * qr; pr = t;
          qr = u_r[2][m1][k1]; qi = u_i[2][m1][k1];
          t  = pr * qr - pi * qi; pi = pr * qi + pi * qr; pr = t;
          qr = u_r[3][m0][k0]; qi = u_i[3][m0][k0];
          t  = pr * qr - pi * qi; pi = pr * qi + pi * qr; pr = t;
          ar[c][v] = pr; ai[c][v] = pi;
        }
      }
      // 4 tiles per wave, 16 waves cover 64 column tiles
      const unsigned ppm = (1u << pp) - 1u;
      const unsigned stride = 1u << pp;
#pragma unroll 1
      for (int it = 0; it < NTILES / NWAVES; ++it) {
        const int t = wv * (NTILES / NWAVES) + it;
        const unsigned colid = (unsigned)(t * 16 + (lane & 15));
        const unsigned bcol  = ((colid >> pp) << (pp + 4)) | (colid & ppm);
        v8f dre  = {0.f, 0.f, 0.f, 0.f, 0.f, 0.f, 0.f, 0.f};
        v8f dim_ = {0.f, 0.f, 0.f, 0.f, 0.f, 0.f, 0.f, 0.f};
#pragma unroll
        for (int c = 0; c < 4; ++c) {
          const unsigned a0 = bcol + (unsigned)(4 * c + 2 * h) * stride;
          const unsigned a1 = a0 + stride;
          v2f br = {s_re[a0], s_re[a1]};
          v2f bi = {s_im[a0], s_im[a1]};
          v2f an = {-ai[c][0], -ai[c][1]};
          // D_re += M_re*X_re + (-M_im)*X_im ; D_im += M_im*X_re + M_re*X_im
          dre  = __builtin_amdgcn_wmma_f32_16x16x4_f32(false, ar[c], false, br, (short)0, dre,  false, false);
          dre  = __builtin_amdgcn_wmma_f32_16x16x4_f32(false, an,    false, bi, (short)0, dre,  false, false);
          dim_ = __builtin_amdgcn_wmma_f32_16x16x4_f32(false, ai[c], false, br, (short)0, dim_, false, false);
          dim_ = __builtin_amdgcn_wmma_f32_16x16x4_f32(false, ar[c], false, bi, (short)0, dim_, false, false);
        }
#pragma unroll
        for (int v = 0; v < 8; ++v) {  // C/D layout: M = v + 8*h, N = lane&15
          const unsigned addr = bcol + (unsigned)(v + 8 * h) * stride;
          s_re[addr] = dre[v];
          s_im[addr] = dim_[v];
        }
      }
      __syncthreads();
    }
#else
    // ---- fallback: 14 sequential 1-qubit gates on VALU ----
    for (int i = 0; i < NUM_WIRES; ++i) {
      const float th = params[layer * 2 * NUM_WIRES + i];
      const float ph = params[layer * 2 * NUM_WIRES + NUM_WIRES + i];
      float s, c, sp, cp;
      __sincosf(th * 0.5f, &s, &c);
      __sincosf(ph * 0.5f, &sp, &cp);
      const float A = c * cp, B2 = c * sp, C2 = s * cp, D2 = s * sp;
      const int pt = NUM_WIRES - 1 - i;
      const unsigned L = 1u << pt;
      for (int p = tid; p < PAIRS; p += NTHREADS) {
        const unsigned i0 = (((unsigned)p >> pt) << (pt + 1)) | ((unsigned)p & (L - 1u));
        const unsigned i1 = i0 + L;
        const float a0r = s_re[i0], a0i = s_im[i0];
        const float a1r = s_re[i1], a1i = s_im[i1];
        s_re[i0] =  A * a0r + B2 * a0i - C2 * a1r - D2 * a1i;
        s_im[i0] = -B2 * a0r + A * a0i + D2 * a1r - C2 * a1i;
        s_re[i1] =  C2 * a0r - D2 * a0i + A * a1r - B2 * a1i;
        s_im[i1] =  D2 * a0r + C2 * a0i + B2 * a1r + A * a1i;
      }
      __syncthreads();
    }
#endif
    // ---- ring of CNOTs ----
#pragma unroll 1
    for (int i = 0; i < NUM_WIRES; ++i) {
      const int pc  = NUM_WIRES - 1 - i;                       // control bit pos
      const int ptb = NUM_WIRES - 1 - ((i + 1) % NUM_WIRES);   // target bit pos
      const int lo = pc < ptb ? pc : ptb;
      const int hi = pc < ptb ? ptb : pc;
      const unsigned blo = (lo == pc) ? 1u : 0u;  // control=1, target=0 enumerated
      const unsigned bhi = (hi == pc) ? 1u : 0u;
      for (int q = tid; q < DIMS / 4; q += NTHREADS) {
        const unsigned x = insert_bit(insert_bit((unsigned)q, lo, blo), hi, bhi);
        const unsigned y = x ^ (1u << ptb);
        float t;
        t = s_re[x]; s_re[x] = s_re[y]; s_re[y] = t;
        t = s_im[x]; s_im[x] = s_im[y]; s_im[y] = t;
      }
      __syncthreads();
    }
  }

  // ---- <Z_w> features + linear head ----
  if (tid < NUM_WIRES) s_red[tid] = 0.f;
  __syncthreads();
  float zacc[NUM_WIRES];
#pragma unroll
  for (int w = 0; w < NUM_WIRES; ++w) zacc[w] = 0.f;
  for (int x = tid; x < DIMS; x += NTHREADS) {
    const float re = s_re[x], im = s_im[x];
    const float p = re * re + im * im;
#pragma unroll
    for (int w = 0; w < NUM_WIRES; ++w) {
      const unsigned bit = ((unsigned)x >> (NUM_WIRES - 1 - w)) & 1u;
      zacc[w] += bit ? -p : p;
    }
  }
#pragma unroll
  for (int w = 0; w < NUM_WIRES; ++w) atomicAdd(&s_red[w], zacc[w]);
  __syncthreads();
  if (tid == 0) {
    float o = head_b[0];
#pragma unroll
    for (int w = 0; w < NUM_WIRES; ++w) o += s_red[w] * head_w[w];
    out[b] = o;
  }
}

extern "C" void kernel_launch(void* const* d_in, const int* in_sizes, int n_in,
                              void* d_out, int out_size, void* d_ws, size_t ws_size,
                              hipStream_t stream) {
  const float* state  = (const float*)d_in[0];
  const float* params = (const float*)d_in[1];
  const float* head_w = (const float*)d_in[2];
  const float* head_b = (const float*)d_in[3];
  float* out = (float*)d_out;
  const int batch = in_sizes[0] / DIMS;   // 512
  qsim_kernel<<<dim3(batch), dim3(NTHREADS), 0, stream>>>(state, params, head_w, head_b, out);
}